// Quantise_32298154066344
// MI455X (gfx1250) — compile-verified
//
#include <hip/hip_runtime.h>
#include <math.h>

// ---------------------------------------------------------------------------
// VQ-VAE quantise for MI455X (gfx1250, wave32, WMMA).
//   input     [32,4096,64] f32   -> 131072 rows of D=64
//   embedding [512,64]     f32   -> M=512 codes
// out = quantised_st (8388608 f32) | commitment | codebook | perplexity
// ---------------------------------------------------------------------------

typedef float v2f __attribute__((ext_vector_type(2)));
typedef float v8f __attribute__((ext_vector_type(8)));

#define BN 131072        // B*N rows
#define DD 64            // feature dim
#define MM 512           // codebook size
#define LDS_STRIDE 68    // padded row stride in floats (68%64=4 -> no bank conflicts)
#define LDS_FLOATS (MM * LDS_STRIDE + MM)        // emb tiles + norms
#define LDS_BYTES (LDS_FLOATS * 4)               // 141312 B dynamic LDS

// ---------------------------------------------------------------------------
// Kernel 1: fused  score = ||e||^2 + (-2x).e  via V_WMMA_F32_16X16X4_F32,
// running argmin over 512 codes. Each wave owns TWO 16-row tiles (32 rows),
// so every LDS B-tile load feeds two WMMAs. 8 waves/block -> 256 rows/block.
// ---------------------------------------------------------------------------
__global__ __launch_bounds__(256) void vq_argmin_kernel(
    const float* __restrict__ x, const float* __restrict__ emb,
    int* __restrict__ idx_out)
{
  extern __shared__ float lds[];
  float* enorms = lds + MM * LDS_STRIDE;
  const int tid = threadIdx.x;

  // Stage full embedding (512x64 f32) into padded LDS; thread t copies rows
  // 2t, 2t+1 and computes ||e||^2 on the fly (norms cached in LDS).
  #pragma unroll
  for (int r = 0; r < 2; ++r) {
    const int row = tid * 2 + r;
    const float4* src = reinterpret_cast<const float4*>(emb + (size_t)row * DD);
    float4* dst = reinterpret_cast<float4*>(&lds[row * LDS_STRIDE]);  // 272B stride, 16B aligned
    float nrm = 0.f;
    #pragma unroll
    for (int v = 0; v < 16; ++v) {
      float4 q = src[v];
      dst[v] = q;
      nrm += q.x * q.x + q.y * q.y + q.z * q.z + q.w * q.w;
    }
    enorms[row] = nrm;
  }
  __syncthreads();

  const int lane   = tid & 31;
  const int wave   = tid >> 5;
  const int lane15 = lane & 15;
  const int half   = lane >> 4;                 // 0: K%4 in {0,1}, 1: K%4 in {2,3}
  const int r0     = (blockIdx.x * 8 + wave) * 32;   // two 16-row tiles

  // A-tiles (2 x 16x64) in registers per the 32-bit A 16x4 layout, scaled by
  // -2 (exact) so score = ||e||^2 + A.e  with no per-tile multiply.
  v2f a0[16], a1[16];
  {
    const float* x0 = x + (size_t)(r0 + lane15) * DD + 2 * half;
    const float* x1 = x0 + 16 * DD;
    #pragma unroll
    for (int j = 0; j < 16; ++j) {
      v2f t0 = *reinterpret_cast<const v2f*>(x0 + 4 * j);
      v2f t1 = *reinterpret_cast<const v2f*>(x1 + 4 * j);
      a0[j].x = -2.0f * t0.x; a0[j].y = -2.0f * t0.y;
      a1[j].x = -2.0f * t1.x; a1[j].y = -2.0f * t1.y;
    }
  }

  float best0[8], best1[8];
  int   bidx0[8], bidx1[8];
  #pragma unroll
  for (int r = 0; r < 8; ++r) {
    best0[r] = 3.4e38f; bidx0[r] = 0;
    best1[r] = 3.4e38f; bidx1[r] = 0;
  }

  for (int c = 0; c < 32; ++c) {                // 32 column tiles of 16 codes
    const int n = c * 16 + lane15;              // this lane's candidate column
    const float* bp = &lds[n * LDS_STRIDE + 2 * half];
    const float enorm = enorms[n];              // ds_load_b32 (broadcast in half 1)
    v8f acc0 = {}, acc1 = {};
    #pragma unroll
    for (int j = 0; j < 16; ++j) {
      v2f b = *reinterpret_cast<const v2f*>(bp + 4 * j);   // one B feeds two WMMAs
      acc0 = __builtin_amdgcn_wmma_f32_16x16x4_f32(
          false, a0[j], false, b, (short)0, acc0, false, false);
      acc1 = __builtin_amdgcn_wmma_f32_16x16x4_f32(
          false, a1[j], false, b, (short)0, acc1, false, false);
    }
    #pragma unroll
    for (int r = 0; r < 8; ++r) {               // C layout: vgpr r = row r (+8 for half 1)
      const float s0 = enorm + acc0[r];
      const float s1 = enorm + acc1[r];
      if (s0 < best0[r]) { best0[r] = s0; bidx0[r] = n; }  // strict: first min wins
      if (s1 < best1[r]) { best1[r] = s1; bidx1[r] = n; }
    }
  }

  // Reduce (min,idx) across the 16 lanes of each half; smaller index on ties.
  #pragma unroll
  for (int m = 1; m <= 8; m <<= 1) {
    #pragma unroll
    for (int r = 0; r < 8; ++r) {
      float ov; int oi;
      ov = __shfl_xor(best0[r], m, 32); oi = __shfl_xor(bidx0[r], m, 32);
      if (ov < best0[r] || (ov == best0[r] && oi < bidx0[r])) { best0[r] = ov; bidx0[r] = oi; }
      ov = __shfl_xor(best1[r], m, 32); oi = __shfl_xor(bidx1[r], m, 32);
      if (ov < best1[r] || (ov == best1[r] && oi < bidx1[r])) { best1[r] = ov; bidx1[r] = oi; }
    }
  }
  if (lane15 == 0) {
    #pragma unroll
    for (int r = 0; r < 8; ++r) {
      idx_out[r0 +      half * 8 + r] = bidx0[r];
      idx_out[r0 + 16 + half * 8 + r] = bidx1[r];
    }
  }
}

// ---------------------------------------------------------------------------
// Kernel 2: gather + straight-through output + SSE partials + histogram.
// 32 rows/block, 8 threads/row, 8 floats/thread.
// ---------------------------------------------------------------------------
__global__ __launch_bounds__(256) void vq_gather_kernel(
    const float* __restrict__ x, const float* __restrict__ emb,
    const int* __restrict__ idx, float* __restrict__ out,
    unsigned* __restrict__ hist, float* __restrict__ partials)
{
  __shared__ float red[256];
  const int tid  = threadIdx.x;
  const int row  = blockIdx.x * 32 + (tid >> 3);
  const int cb   = (tid & 7) * 8;
  const int code = idx[row];
  const float* e  = emb + (size_t)code * DD + cb;
  const float* xi = x   + (size_t)row  * DD + cb;
  float*       o  = out + (size_t)row  * DD + cb;

  float sse = 0.f;
  #pragma unroll
  for (int v = 0; v < 2; ++v) {
    float4 q  = reinterpret_cast<const float4*>(e)[v];
    float4 xv = reinterpret_cast<const float4*>(xi)[v];
    float4 ov;
    float d;
    d = xv.x - q.x; sse += d * d; ov.x = xv.x + (q.x - xv.x);  // match x + sg(q - x)
    d = xv.y - q.y; sse += d * d; ov.y = xv.y + (q.y - xv.y);
    d = xv.z - q.z; sse += d * d; ov.z = xv.z + (q.z - xv.z);
    d = xv.w - q.w; sse += d * d; ov.w = xv.w + (q.w - xv.w);
    reinterpret_cast<float4*>(o)[v] = ov;
  }
  if ((tid & 7) == 0) atomicAdd(&hist[code], 1u);

  red[tid] = sse;
  __syncthreads();
  for (int s = 128; s > 0; s >>= 1) {
    if (tid < s) red[tid] += red[tid + s];
    __syncthreads();
  }
  if (tid == 0) partials[blockIdx.x] = red[0];
}

// ---------------------------------------------------------------------------
// Kernel 3: final scalars. One block of 512 threads.
// ---------------------------------------------------------------------------
__global__ __launch_bounds__(512) void vq_finalize_kernel(
    const float* __restrict__ partials, const unsigned* __restrict__ hist,
    float* __restrict__ scal)
{
  __shared__ float red[512];
  const int tid = threadIdx.x;

  float s = 0.f;
  #pragma unroll
  for (int k = 0; k < 8; ++k) s += partials[tid + 512 * k];   // 4096 partials
  red[tid] = s;
  __syncthreads();
  for (int st = 256; st > 0; st >>= 1) {
    if (tid < st) red[tid] += red[tid + st];
    __syncthreads();
  }
  const float sse = red[0];
  __syncthreads();

  const float avg = (float)hist[tid] / (float)BN;             // tid < 512 == MM
  red[tid] = avg * logf(avg + 1e-10f);
  __syncthreads();
  for (int st = 256; st > 0; st >>= 1) {
    if (tid < st) red[tid] += red[tid + st];
    __syncthreads();
  }
  if (tid == 0) {
    const float mse = sse / (float)(BN * DD);
    scal[0] = 0.25f * mse;        // commitment_loss
    scal[1] = mse;                // codebook_loss
    scal[2] = expf(-red[0]);      // perplexity
  }
}

__global__ void vq_zero_hist(unsigned* __restrict__ hist) {
  hist[threadIdx.x] = 0u;
}

// ---------------------------------------------------------------------------
extern "C" void kernel_launch(void* const* d_in, const int* in_sizes, int n_in,
                              void* d_out, int out_size, void* d_ws, size_t ws_size,
                              hipStream_t stream) {
  const float* x   = (const float*)d_in[0];
  const float* emb = (const float*)d_in[1];
  float* out = (float*)d_out;

  // workspace layout: indices | histogram | partials
  int*      idx      = (int*)d_ws;
  unsigned* hist     = (unsigned*)((char*)d_ws + (size_t)BN * sizeof(int));
  float*    partials = (float*)((char*)d_ws + (size_t)BN * sizeof(int) + MM * sizeof(unsigned));
  float*    scal     = out + (size_t)BN * DD;

  static_assert(LDS_BYTES == 141312, "lds size");
  hipFuncSetAttribute((const void*)vq_argmin_kernel,
                      hipFuncAttributeMaxDynamicSharedMemorySize, LDS_BYTES);

  vq_zero_hist<<<1, MM, 0, stream>>>(hist);
  vq_argmin_kernel<<<BN / 256, 256, LDS_BYTES, stream>>>(x, emb, idx);      // 512 blocks
  vq_gather_kernel<<<BN / 32, 256, 0, stream>>>(x, emb, idx, out, hist, partials); // 4096 blocks
  vq_finalize_kernel<<<1, 512, 0, stream>>>(partials, hist, scal);
}